// GNN_62285615727516
// MI455X (gfx1250) — compile-verified
//
#include <hip/hip_runtime.h>

#define N_NODES 100000
#define N_PAD   100032        // multiple of 64 (GEMM row tile); padded node rows
#define IN_CH   128
#define C1      128           // HEADS*HID layer1 output width
#define C2      64            // layer2 output width
#define H1      2
#define H2      1
#define NEDGES  800000
#define EDG     (NEDGES + N_NODES)   // with self loops

typedef __attribute__((ext_vector_type(2))) float v2f;
typedef __attribute__((ext_vector_type(8))) float v8f;

// ---------- helpers ----------
__device__ __forceinline__ void edge_sd(const int* ei, int e, int& s, int& d) {
    if (e < NEDGES) { s = ei[e]; d = ei[NEDGES + e]; }
    else            { s = e - NEDGES; d = s; }
}
// order-preserving float<->uint key (atomic max on floats incl. negatives)
__device__ __forceinline__ unsigned fkey(float f) {
    unsigned b = __float_as_uint(f);
    return (b & 0x80000000u) ? ~b : (b | 0x80000000u);
}
__device__ __forceinline__ float funkey(unsigned k) {
    return __uint_as_float((k & 0x80000000u) ? (k ^ 0x80000000u) : ~k);
}
#define KEY_NEG_INF 0x007FFFFFu   // fkey(-inf)

// ---------- tiny fill kernels (scratch is poisoned each run) ----------
__global__ void fill_f32(float* p, float v, int n) {
    int i = blockIdx.x * 256 + threadIdx.x;
    if (i < n) p[i] = v;
}
__global__ void fill_u32(unsigned* p, unsigned v, int n) {
    int i = blockIdx.x * 256 + threadIdx.x;
    if (i < n) p[i] = v;
}

// ---------- h0 = emb * (1 + x) ----------
__global__ void prep_kernel(const float* __restrict__ x,
                            const float* __restrict__ emb,
                            float* __restrict__ h0, int n) {
    int i = blockIdx.x * 256 + threadIdx.x;
    if (i < n) h0[i] = emb[i] * (1.0f + x[i]);
}

// ---------- WMMA GEMM: Y[N_PAD,OC] = X[N_PAD,128] @ W[128,OC] ----------
// 256 threads (8 waves), 64 rows/block. X tile in LDS (stride 132 => the 16
// A-fragment lanes hit 16 distinct banks). Each wave owns ONE column tile and
// OC/32 row tiles: B fragments loaded once per k and reused across row tiles.
// Buffers padded to N_PAD rows -> unconditional stores, pure 32-bit indexing.
#define XS_STRIDE 132
template <int OC>
__global__ void gemm_wmma_kernel(const float* __restrict__ X,
                                 const float* __restrict__ W,
                                 float* __restrict__ Y) {
    constexpr int NCT = OC / 16;   // column tiles (8 or 4)
    constexpr int MPW = OC / 32;   // row tiles per wave (4 or 2)
    __shared__ float Xs[64 * XS_STRIDE];

    const int tid  = threadIdx.x;
    const int wave = tid >> 5;
    const int lane = tid & 31;
    const int rowbase = blockIdx.x * 64;

    // cooperative coalesced load of 64x128 X tile
    #pragma unroll
    for (int it = 0; it < 32; ++it) {
        int idx = it * 256 + tid;
        int r = idx >> 7, c = idx & 127;
        Xs[r * XS_STRIDE + c] = X[(rowbase + r) * IN_CH + c];
    }
    __syncthreads();

    const int n    = wave % NCT;              // column tile
    const int m0   = (wave / NCT) * MPW;      // first row tile
    const int laneM = lane & 15;
    const int koff  = (lane >> 4) * 2;        // 0 or 2
    const int col   = n * 16 + laneM;

    const float* Wp = W + koff * OC + col;    // immediate offsets cover k-loop
    const float* xb[MPW];
    v8f acc[MPW];
    #pragma unroll
    for (int m = 0; m < MPW; ++m) {
        xb[m] = &Xs[((m0 + m) * 16 + laneM) * XS_STRIDE + koff];
        acc[m] = (v8f){0.f, 0.f, 0.f, 0.f, 0.f, 0.f, 0.f, 0.f};
    }

    #pragma unroll
    for (int k = 0; k < 32; ++k) {
        const int kk = k * 4;
        v2f b;                                 // B 4x16: K = kk+koff+{0,1}
        b.x = Wp[kk * OC];
        b.y = Wp[(kk + 1) * OC];
        #pragma unroll
        for (int m = 0; m < MPW; ++m) {
            v2f a = *(const v2f*)(xb[m] + kk); // A 16x4: row=laneM, K=kk+koff+{0,1}
            acc[m] = __builtin_amdgcn_wmma_f32_16x16x4_f32(
                false, a, false, b, (short)0, acc[m], false, false);
        }
    }

    const int half = (lane >> 4) ? 8 : 0;      // D: VGPR r -> row r (+8 upper half)
    #pragma unroll
    for (int m = 0; m < MPW; ++m) {
        float* yp = Y + (rowbase + (m0 + m) * 16 + half) * OC + col;
        #pragma unroll
        for (int r = 0; r < 8; ++r) yp[r * OC] = acc[m][r];
    }
}

// ---------- per-node attention logits: al[n,h] = sum_c h[n,c]*a[c], head=c>>6 ----------
template <int C, int H>
__global__ void al_kernel(const float* __restrict__ h,
                          const float* __restrict__ a_src,
                          const float* __restrict__ a_dst,
                          float* __restrict__ als, float* __restrict__ ald) {
    const int wave = threadIdx.x >> 5, lane = threadIdx.x & 31;
    const int nnode = blockIdx.x * 8 + wave;
    if (nnode >= N_NODES) return;
    float ps[2] = {0.f, 0.f}, pd[2] = {0.f, 0.f};
    #pragma unroll
    for (int i = 0; i < C / 32; ++i) {
        int c = lane + 32 * i;
        float hv = h[nnode * C + c];
        int hd = c >> 6;                      // HID == 64 in both layers
        ps[hd] += hv * a_src[c];
        pd[hd] += hv * a_dst[c];
    }
    #pragma unroll
    for (int off = 16; off; off >>= 1) {
        ps[0] += __shfl_xor(ps[0], off, 32);
        pd[0] += __shfl_xor(pd[0], off, 32);
        if (H > 1) {
            ps[1] += __shfl_xor(ps[1], off, 32);
            pd[1] += __shfl_xor(pd[1], off, 32);
        }
    }
    if (lane == 0) {
        #pragma unroll
        for (int hh = 0; hh < H; ++hh) {
            als[nnode * H + hh] = ps[hh];
            ald[nnode * H + hh] = pd[hh];
        }
    }
}

// ---------- edge pass 1: e = leakyrelu02(al_src[s]+al_dst[d]); atomic max per dst ----------
template <int H>
__global__ void edge_logits_max(const float* __restrict__ als,
                                const float* __restrict__ ald,
                                const int* __restrict__ ei,
                                float* __restrict__ ebuf,
                                unsigned* __restrict__ mkey) {
    int e = blockIdx.x * 256 + threadIdx.x;
    if (e >= EDG) return;
    int s, d; edge_sd(ei, e, s, d);
    #pragma unroll
    for (int h = 0; h < H; ++h) {
        float v = als[s * H + h] + ald[d * H + h];
        v = (v > 0.f) ? v : 0.2f * v;
        ebuf[e * H + h] = v;
        atomicMax(&mkey[d * H + h], fkey(v));
    }
}

// ---------- edge pass 2: ex = exp(e - m[dst]); denom[dst] += ex ----------
template <int H>
__global__ void edge_exp_sum(const int* __restrict__ ei,
                             float* __restrict__ ebuf,
                             const unsigned* __restrict__ mkey,
                             float* __restrict__ denom) {
    int e = blockIdx.x * 256 + threadIdx.x;
    if (e >= EDG) return;
    int s, d; edge_sd(ei, e, s, d);
    #pragma unroll
    for (int h = 0; h < H; ++h) {
        float m = funkey(mkey[d * H + h]);
        float ex = __expf(ebuf[e * H + h] - m);
        ebuf[e * H + h] = ex;
        atomicAdd(&denom[d * H + h], ex);
    }
}

// ---------- edge pass 3: acc[dst,c] += (ex/denom[dst,head]) * h[src,c]  (wave/edge) ----------
template <int C, int H>
__global__ void edge_scatter(const int* __restrict__ ei,
                             const float* __restrict__ h,
                             const float* __restrict__ ebuf,
                             const float* __restrict__ denom,
                             float* __restrict__ acc) {
    const int wave = threadIdx.x >> 5, lane = threadIdx.x & 31;
    int e = blockIdx.x * 8 + wave;
    if (e >= EDG) return;
    int s, d; edge_sd(ei, e, s, d);
    float a0 = ebuf[e * H] / denom[d * H];
    float a1 = (H > 1) ? (ebuf[e * H + 1] / denom[d * H + 1]) : a0;
    #pragma unroll
    for (int i = 0; i < C / 32; ++i) {
        int c = lane + 32 * i;
        float alpha = (c >> 6) ? a1 : a0;
        atomicAdd(&acc[d * C + c], alpha * h[s * C + c]);
    }
}

// ---------- finalize in place: v = leakyrelu001(acc + bias[c]) ----------
__global__ void finalize_kernel(float* __restrict__ acc,
                                const float* __restrict__ bias, int n, int Cmask) {
    int i = blockIdx.x * 256 + threadIdx.x;
    if (i >= n) return;
    float v = acc[i] + bias[i & Cmask];
    acc[i] = (v > 0.f) ? v : 0.01f * v;
}

// ---------------------------------------------------------------------------
extern "C" void kernel_launch(void* const* d_in, const int* in_sizes, int n_in,
                              void* d_out, int out_size, void* d_ws, size_t ws_size,
                              hipStream_t stream) {
    const float* x      = (const float*)d_in[0];
    const int*   ei     = (const int*)  d_in[1];
    const float* emb    = (const float*)d_in[2];
    const float* W1     = (const float*)d_in[3];
    const float* a_src1 = (const float*)d_in[4];
    const float* a_dst1 = (const float*)d_in[5];
    const float* b1     = (const float*)d_in[6];
    const float* W2     = (const float*)d_in[7];
    const float* a_src2 = (const float*)d_in[8];
    const float* a_dst2 = (const float*)d_in[9];
    const float* b2     = (const float*)d_in[10];
    float* out = (float*)d_out;
    (void)in_sizes; (void)n_in; (void)out_size; (void)ws_size;

    // ---- workspace carve-up (feature buffers padded to N_PAD rows) ----
    char* p = (char*)d_ws;
    float*    h0    = (float*)p;    p += (size_t)N_PAD * C1 * 4;   // doubles as acc1
    float*    hT    = (float*)p;    p += (size_t)N_PAD * C1 * 4;   // h1 then h2
    float*    als1  = (float*)p;    p += (size_t)N_NODES * H1 * 4;
    float*    ald1  = (float*)p;    p += (size_t)N_NODES * H1 * 4;
    unsigned* mkey1 = (unsigned*)p; p += (size_t)N_NODES * H1 * 4;
    float*    den1  = (float*)p;    p += (size_t)N_NODES * H1 * 4;
    float*    ebuf1 = (float*)p;    p += (size_t)EDG     * H1 * 4;
    float*    als2  = (float*)p;    p += (size_t)N_NODES * 4;
    float*    ald2  = (float*)p;    p += (size_t)N_NODES * 4;
    unsigned* mkey2 = (unsigned*)p; p += (size_t)N_NODES * 4;
    float*    den2  = (float*)p;    p += (size_t)N_NODES * 4;
    float*    ebuf2 = (float*)p;    p += (size_t)EDG * 4;

    const int NE  = N_NODES * C1;   // 12.8M
    const int NE2 = N_NODES * C2;   // 6.4M
    dim3 blk(256);
    auto cdiv = [](int a, int b) { return (a + b - 1) / b; };
    const int gemmBlocks = N_PAD / 64;   // 1563

    // ---------- stage 0: h0 = emb*(1+x) ----------
    prep_kernel<<<cdiv(NE, 256), blk, 0, stream>>>(x, emb, h0, NE);

    // ---------- layer 1 ----------
    gemm_wmma_kernel<C1><<<gemmBlocks, blk, 0, stream>>>(h0, W1, hT);
    al_kernel<C1, H1><<<cdiv(N_NODES, 8), blk, 0, stream>>>(hT, a_src1, a_dst1, als1, ald1);
    fill_u32<<<cdiv(N_NODES * H1, 256), blk, 0, stream>>>(mkey1, KEY_NEG_INF, N_NODES * H1);
    fill_f32<<<cdiv(N_NODES * H1, 256), blk, 0, stream>>>(den1, 0.f, N_NODES * H1);
    fill_f32<<<cdiv(NE, 256), blk, 0, stream>>>(h0, 0.f, NE);   // h0 now acc1
    edge_logits_max<H1><<<cdiv(EDG, 256), blk, 0, stream>>>(als1, ald1, ei, ebuf1, mkey1);
    edge_exp_sum<H1><<<cdiv(EDG, 256), blk, 0, stream>>>(ei, ebuf1, mkey1, den1);
    edge_scatter<C1, H1><<<cdiv(EDG, 8), blk, 0, stream>>>(ei, hT, ebuf1, den1, h0);
    finalize_kernel<<<cdiv(NE, 256), blk, 0, stream>>>(h0, b1, NE, C1 - 1);

    // ---------- layer 2 ----------
    gemm_wmma_kernel<C2><<<gemmBlocks, blk, 0, stream>>>(h0, W2, hT);
    al_kernel<C2, H2><<<cdiv(N_NODES, 8), blk, 0, stream>>>(hT, a_src2, a_dst2, als2, ald2);
    fill_u32<<<cdiv(N_NODES, 256), blk, 0, stream>>>(mkey2, KEY_NEG_INF, N_NODES);
    fill_f32<<<cdiv(N_NODES, 256), blk, 0, stream>>>(den2, 0.f, N_NODES);
    fill_f32<<<cdiv(NE2, 256), blk, 0, stream>>>(out, 0.f, NE2);
    edge_logits_max<H2><<<cdiv(EDG, 256), blk, 0, stream>>>(als2, ald2, ei, ebuf2, mkey2);
    edge_exp_sum<H2><<<cdiv(EDG, 256), blk, 0, stream>>>(ei, ebuf2, mkey2, den2);
    edge_scatter<C2, H2><<<cdiv(EDG, 8), blk, 0, stream>>>(ei, hT, ebuf2, den2, out);
    finalize_kernel<<<cdiv(NE2, 256), blk, 0, stream>>>(out, b2, NE2, C2 - 1);
}